// simplex_frame_84731114816063
// MI455X (gfx1250) — compile-verified
//
#include <hip/hip_runtime.h>
#include <stdint.h>

#define N_USERS    100000
#define N_ITEMS    50000
#define N_TOTAL    150000
#define DIM        64
#define NNZ_E      2000000
#define N_HOPS     3
#define OUT_STRIDE ((N_HOPS + 1) * DIM)   // 256 floats per node row of out[N,4,64]

typedef __attribute__((ext_vector_type(2))) float v2f;
typedef __attribute__((ext_vector_type(8))) float v8f;

// ---------------- deterministic per-call RNG hash ----------------
__device__ __forceinline__ uint32_t mix32(uint32_t a, uint32_t b) {
  uint32_t h = a * 0x9E3779B9u + 0x7F4A7C15u;
  h ^= b + 0x85EBCA6Bu + (h << 6) + (h >> 2);
  h ^= h >> 16; h *= 0x7FEB352Du;
  h ^= h >> 15; h *= 0x846CA68Bu;
  h ^= h >> 16;
  return h;
}
__device__ __forceinline__ float u01(uint32_t h) {
  return (float)(h >> 8) * (1.0f / 16777216.0f);
}
// post-dropout edge weight (keep iff u >= 1-EDGE_RATE(0.5); scale 1/(1-rate)=2)
__device__ __forceinline__ float edge_w(const float* evals, long long e,
                                        int hop, uint32_t seed) {
  float v = evals[e];
  uint32_t h = mix32(seed + (uint32_t)hop * 0x632BE5ABu, (uint32_t)e);
  return (u01(h) >= 0.5f) ? v * 2.0f : 0.0f;
}

// ---------------- zero the full output ----------------
__global__ void zero_out_kernel(float4* __restrict__ out, long long n4) {
  long long i = (long long)blockIdx.x * blockDim.x + threadIdx.x;
  if (i < n4) out[i] = make_float4(0.f, 0.f, 0.f, 0.f);
}

// ---------------- hop-0 copy via CDNA5 async LDS data mover ----------------
// out[:,0,:] = concat(user_embed, item_embed); dest rows are 256B at 1024B stride.
__global__ __launch_bounds__(256)
void copy_hop0_kernel(const float* __restrict__ user_e,
                      const float* __restrict__ item_e,
                      float* __restrict__ out) {
  __shared__ __align__(16) float staging[256 * 4];
  const int t = threadIdx.x;
  long long v = (long long)blockIdx.x * blockDim.x + t;     // one 16B packet per lane
  const long long nVec = (long long)N_TOTAL * DIM / 4;      // 2.4M packets
  uint32_t ldsOff = (uint32_t)(uintptr_t)&staging[t * 4];   // low 32b of shared ptr = LDS byte offset
  if (v < nVec) {
    long long e    = v * 4;
    long long node = e >> 6;
    int       d    = (int)(e & 63);
    const float* src = (node < N_USERS)
                     ? user_e + node * DIM + d
                     : item_e + (node - N_USERS) * DIM + d;
    const float* dst = out + node * OUT_STRIDE + d;
    asm volatile("global_load_async_to_lds_b128 %0, %1, off"
                 :: "v"(ldsOff), "v"(src) : "memory");
    asm volatile("s_wait_asynccnt 0x0" ::: "memory");
    asm volatile("global_store_async_from_lds_b128 %0, %1, off"
                 :: "v"(dst), "v"(ldsOff) : "memory");
    // S_ENDPGM performs an implicit wait-idle for the outstanding async store.
  }
}

// ---------------- SpMM hop: one wave32 per 16-edge batch ----------------
// msg(16x64) = diag(v) * G, G = gathered prev rows, done on the matrix pipe:
// per 16x16 tile:  C = sum_{g=0..3} A_g(16x4) x B_g(4x16),
//   A_g[m][k] = v_m  iff m == 4g+k   (selection => exact fp32 row scaling)
//   B_g[k][n] = prev[col[4g+k]][tile*16+n]
// A_g is tile-invariant and hoisted; gathers/scatters use uniform SGPR bases
// plus 32-bit per-lane offsets (whole out buffer < 2^31 bytes).
__global__ __launch_bounds__(256)
void spmm_hop_kernel(const float* out_base,
                     const float* __restrict__ evals,
                     const int*   __restrict__ erows,
                     const int*   __restrict__ ecols,
                     int hop, uint32_t seed) {
  const int lane = threadIdx.x & 31;
  const int wave = threadIdx.x >> 5;
  const int m    = lane & 15;      // edge index inside batch, also A's M and C's N
  const int half = lane >> 4;      // 0 -> K{0,1}, 1 -> K{2,3} in the A/B VGPR pair
  const int kb   = half * 2;

  long long batch = (long long)blockIdx.x * 8 + wave;
  const long long nBatches = (NNZ_E + 15) / 16;   // 125000, exact (no tail)
  if (batch >= nBatches) return;
  const long long e0 = batch * 16;

  const float* prev = out_base + (long long)(hop - 1) * DIM;   // out[:,hop-1,:] (uniform base)
  float*       cur  = (float*)out_base + (long long)hop * DIM; // out[:,hop,:]   (uniform base)

  // A diagonal value for this lane's own edge (m); tile-invariant A_g hoisted.
  const float vv = edge_w(evals, e0 + m, hop, seed);
  v2f a[4];
  #pragma unroll
  for (int g = 0; g < 4; ++g) {
    const int k0 = 4 * g + kb;
    a[g].x = (m == k0    ) ? vv : 0.0f;   // A VGPR0 : K = kb   (lanes16-31: K=2)
    a[g].y = (m == k0 + 1) ? vv : 0.0f;   // A VGPR1 : K = kb+1 (lanes16-31: K=3)
  }

  // Gather element offsets (u32): B rows K = 4g+kb, 4g+kb+1, column m.
  uint32_t go0[4], go1[4];
  #pragma unroll
  for (int g = 0; g < 4; ++g) {
    const int k0 = 4 * g + kb;
    go0[g] = (uint32_t)ecols[e0 + k0]     * OUT_STRIDE + m;
    go1[g] = (uint32_t)ecols[e0 + k0 + 1] * OUT_STRIDE + m;
  }
  // Scatter byte offsets (u32): C VGPR r holds (M = r + 8*half, N = m).
  uint32_t so[8];
  #pragma unroll
  for (int r = 0; r < 8; ++r)
    so[r] = (uint32_t)erows[e0 + r + 8 * half] * (OUT_STRIDE * 4u) + m * 4u;

  #pragma unroll
  for (int tile = 0; tile < 4; ++tile) {
    v8f acc = {};
    #pragma unroll
    for (int g = 0; g < 4; ++g) {
      v2f b;
      b.x = prev[go0[g] + tile * 16];    // saddr-form gathers, immediate tile offset
      b.y = prev[go1[g] + tile * 16];
      acc = __builtin_amdgcn_wmma_f32_16x16x4_f32(false, a[g], false, b,
                                                  (short)0, acc, false, false);
    }
    #pragma unroll
    for (int r = 0; r < 8; ++r) {
      float val = acc[r];
      if (val != 0.0f) {                 // dropped edges / zero products: skip no-op adds
        uint32_t off = so[r] + tile * 64u;   // bytes
        asm volatile("global_atomic_add_f32 %0, %1, %2 scope:SCOPE_DEV"
                     :: "v"(off), "v"(val), "s"(cur) : "memory");
      }
    }
  }
}

// ---------------- message dropout, in place on out[:,hop,:], float4 wide ----
__global__ void mess_dropout_kernel(float* out_base, int hop, uint32_t seed) {
  long long i = (long long)blockIdx.x * blockDim.x + threadIdx.x;   // one float4 each
  const long long total4 = (long long)N_TOTAL * DIM / 4;            // 2.4M
  if (i >= total4) return;
  long long e    = i * 4;
  long long node = e >> 6;
  int       d    = (int)(e & 63);
  float4* p = (float4*)(out_base + node * OUT_STRIDE + (long long)hop * DIM + d);
  float4 val = *p;
  const uint32_t hs = seed + (uint32_t)hop * 0xB5297A4Du;
  float* f = &val.x;
  #pragma unroll
  for (int j = 0; j < 4; ++j) {
    uint32_t h = mix32(hs, (uint32_t)(e + j) ^ 0x68E31DA4u);
    f[j] = (u01(h) >= 0.1f) ? f[j] * (1.0f / 0.9f) : 0.0f;  // keep prob 0.9
  }
  *p = val;
}

// ---------------- launch ----------------
extern "C" void kernel_launch(void* const* d_in, const int* in_sizes, int n_in,
                              void* d_out, int out_size, void* d_ws, size_t ws_size,
                              hipStream_t stream) {
  (void)in_sizes; (void)n_in; (void)d_ws; (void)ws_size;
  const float* user_e = (const float*)d_in[0];
  const float* item_e = (const float*)d_in[1];
  const float* evals  = (const float*)d_in[2];
  const int*   erows  = (const int*)d_in[3];
  const int*   ecols  = (const int*)d_in[4];
  float* out = (float*)d_out;

  // 1) zero everything (SpMM accumulates with atomics; also re-inits between replays)
  long long n4 = (long long)out_size / 4;   // 9.6M float4
  zero_out_kernel<<<(int)((n4 + 255) / 256), 256, 0, stream>>>((float4*)out, n4);

  // 2) hop-0 slice via async LDS data mover
  copy_hop0_kernel<<<(int)(((long long)N_TOTAL * DIM / 4 + 255) / 256), 256, 0, stream>>>(
      user_e, item_e, out);

  // 3) hops: WMMA SpMM + message dropout, ping-ponging through out slices
  const int spmmBlocks = (int)(((NNZ_E + 15) / 16 * 32 + 255) / 256);       // 15625
  const int dropBlocks = (int)(((long long)N_TOTAL * DIM / 4 + 255) / 256); // 9375
  for (int hop = 1; hop <= N_HOPS; ++hop) {
    spmm_hop_kernel<<<spmmBlocks, 256, 0, stream>>>(out, evals, erows, ecols,
                                                    hop, 0xC0FFEE01u);
    mess_dropout_kernel<<<dropBlocks, 256, 0, stream>>>(out, hop, 0x0DECAF02u);
  }
}